// AffineTransform2D_5033701671586
// MI455X (gfx1250) — compile-verified
//
#include <hip/hip_runtime.h>

// Spatial transformer: bilinear affine resample of [MB, D, H, W, C] fp32.
// MB=8, D=32, H=W=256, C=4.  Bandwidth-bound gather kernel:
//   - one thread per (b, grid point n), looping all 32 depths (weights amortized
//     over D*C = 128 outputs), unrolled x4 for memory-level parallelism
//   - C=4 floats -> native 16B global_load_b128 / global_store_b128
//   - theta loads are wave-uniform (b per blockIdx.y) -> scalar loads
//   - output stored with non-temporal hint (write-once 256 MiB stream must not
//     evict the input corners that neighboring waves re-read from L2)

typedef float v4f __attribute__((ext_vector_type(4)));

#define MB_    8
#define DEPTH_ 32
#define H_     256
#define W_     256
#define HW_    (H_ * W_)

__global__ __launch_bounds__(256, 8)
void affine_bilinear_kernel(const float* __restrict__ im,
                            const float* __restrict__ thetas,
                            float* __restrict__ out)
{
    const int b = blockIdx.y;
    const int n = blockIdx.x * 256 + threadIdx.x;   // 0 .. HW-1
    // Reference meshgrid flattening: n = i*H + j ; X varies with j, Y with i.
    const int i = n >> 8;     // n / 256
    const int j = n & 255;    // n % 256

    // Per-batch affine (uniform across block -> s_load path)
    const float* th = thetas + b * 6;
    const float t0 = th[0], t1 = th[1], t2 = th[2];
    const float t3 = th[3], t4 = th[4], t5 = th[5];

    const float Xg = fmaf((float)j, 2.0f / (H_ - 1), -1.0f);
    const float Yg = fmaf((float)i, 2.0f / (W_ - 1), -1.0f);

    const float Xc = (fmaf(t0, Xg, fmaf(t1, Yg, t2)) + 1.0f) * (0.5f * (float)W_);
    const float Yc = (fmaf(t3, Xg, fmaf(t4, Yg, t5)) + 1.0f) * (0.5f * (float)H_);

    const float fx = floorf(Xc);
    const float fy = floorf(Yc);
    int x0 = (int)fx;  int x1 = x0 + 1;
    int y0 = (int)fy;  int y1 = y0 + 1;
    x0 = min(max(x0, 0), W_ - 1);
    x1 = min(max(x1, 0), W_ - 1);
    y0 = min(max(y0, 0), H_ - 1);
    y1 = min(max(y1, 0), H_ - 1);

    const float x0f = (float)x0, x1f = (float)x1;
    const float y0f = (float)y0, y1f = (float)y1;

    const float wa = (x1f - Xc) * (y1f - Yc);
    const float wb = (x1f - Xc) * (Yc  - y0f);
    const float wc = (Xc  - x0f) * (y1f - Yc);
    const float wd = (Xc  - x0f) * (Yc  - y0f);

    // float4-unit indexing: element (b,d,y,x) lives at (b*D + d)*HW + y*W + x
    const v4f* __restrict__ base = (const v4f*)im + (size_t)b * DEPTH_ * HW_;
    const v4f* __restrict__ pa = base + (y0 * W_ + x0);
    const v4f* __restrict__ pb = base + (y1 * W_ + x0);
    const v4f* __restrict__ pc = base + (y0 * W_ + x1);
    const v4f* __restrict__ pd = base + (y1 * W_ + x1);
    v4f* __restrict__ po = (v4f*)out + (size_t)b * DEPTH_ * HW_ + n;

#pragma unroll 4
    for (int d = 0; d < DEPTH_; ++d) {
        const size_t off = (size_t)d * HW_;
        const v4f A  = pa[off];
        const v4f B  = pb[off];
        const v4f Cc = pc[off];
        const v4f Dd = pd[off];
        v4f r;
        r.x = fmaf(wa, A.x, fmaf(wb, B.x, fmaf(wc, Cc.x, wd * Dd.x)));
        r.y = fmaf(wa, A.y, fmaf(wb, B.y, fmaf(wc, Cc.y, wd * Dd.y)));
        r.z = fmaf(wa, A.z, fmaf(wb, B.z, fmaf(wc, Cc.z, wd * Dd.z)));
        r.w = fmaf(wa, A.w, fmaf(wb, B.w, fmaf(wc, Cc.w, wd * Dd.w)));
        __builtin_nontemporal_store(r, po + off);   // TH_STORE_NT: write-once stream
    }
}

extern "C" void kernel_launch(void* const* d_in, const int* in_sizes, int n_in,
                              void* d_out, int out_size, void* d_ws, size_t ws_size,
                              hipStream_t stream) {
    const float* im     = (const float*)d_in[0];  // [8, 32, 256, 256, 4] fp32
    const float* thetas = (const float*)d_in[1];  // [8, 6] fp32
    float*       out    = (float*)d_out;          // [8, 32, 256, 256, 4] fp32

    dim3 grid(HW_ / 256, MB_);   // (256, 8)
    dim3 block(256);             // 8 wave32 waves per block
    affine_bilinear_kernel<<<grid, block, 0, stream>>>(im, thetas, out);
}